// CRF_65901978190508
// MI455X (gfx1250) — compile-verified
//
#include <hip/hip_runtime.h>
#include <hip/hip_bf16.h>
#include <math.h>

typedef __attribute__((ext_vector_type(2))) float v2f;
typedef __attribute__((ext_vector_type(8))) float v8f;

#define BB 64
#define SS 512
#define HH 1024
#define TT 24
#define KC 128          // K-chunk staged in LDS per double-buffer slot
#define KP (KC + 4)     // padded row stride (floats) to avoid LDS bank conflicts

// ---------------------------------------------------------------------------
// Kernel 1: emission[m, t] = feats[m, :] . W[t, :] + bias[t],  m = b*S + s
// GEMM M=32768, N=24 (padded to 32), K=1024 via V_WMMA_F32_16X16X4_F32.
// W is staged in LDS with double-buffered GLOBAL_LOAD_ASYNC_TO_LDS_B128
// (ASYNCcnt), rows 24..31 pre-zeroed so the N-pad needs no per-iter select.
// feats streams from HBM (the bandwidth-bound stream) with prefetch.
// ---------------------------------------------------------------------------
__global__ __launch_bounds__(256) void crf_emission_wmma(
    const float* __restrict__ feats, const float* __restrict__ W,
    const float* __restrict__ bias, float* __restrict__ emission) {
  __shared__ float sW[2][32][KP];

  const int tid  = threadIdx.x;
  const int lane = tid & 31;
  const int wave = tid >> 5;
  const int m0   = (blockIdx.x * 8 + wave) * 16;
  const int half = lane >> 4;            // 0: K pair {0,1}, 1: K pair {2,3}
  const int l16  = lane & 15;
  const int koff = half * 2;

  const int rowA = m0 + l16;
  const int col0 = l16;                  // N-tile 0: columns 0..15
  const int col1 = 16 + l16;             // N-tile 1: columns 16..31 (24..31 zero)

  const float* aptr = feats + (size_t)rowA * HH + koff;

  // zero the pad rows 24..31 of both buffers (async loads never touch them)
  for (int i = tid; i < 2 * 8 * KC; i += 256) {
    const int buf = i / (8 * KC);
    const int rem = i % (8 * KC);
    sW[buf][24 + rem / KC][rem % KC] = 0.0f;
  }
  __syncthreads();

  // issue async staging of one W chunk (24 rows x KC floats) into LDS buffer
  auto load_chunk = [&](int buf, int kk) {
    // 24 * (KC/4) = 768 x b128 transfers, 3 per thread
    for (int i = tid; i < 24 * (KC / 4); i += 256) {
      const int row = i / (KC / 4);
      const int c4  = (i % (KC / 4)) * 4;
      const unsigned lds = (unsigned)(uintptr_t)&sW[buf][row][c4];
      const float* g = W + (size_t)row * HH + kk + c4;
      asm volatile("global_load_async_to_lds_b128 %0, %1, off"
                   :: "v"(lds), "v"(g) : "memory");
    }
  };

  load_chunk(0, 0);

  v8f acc0 = {};
  v8f acc1 = {};

  for (int c = 0; c < HH / KC; ++c) {
    const int cur = c & 1;
    asm volatile("s_wait_asynccnt 0x0" ::: "memory");
    __syncthreads();                      // buffer `cur` fully staged for all waves
    if (c + 1 < HH / KC) load_chunk(cur ^ 1, (c + 1) * KC);

    const float* a_c = aptr + c * KC;
    __builtin_prefetch(a_c + KC, 0, 1);   // speculative next feats chunk
#pragma unroll 8
    for (int k4 = 0; k4 < KC; k4 += 4) {
      v2f a  = *(const v2f*)(a_c + k4);
      v2f b0 = *(const v2f*)(&sW[cur][col0][k4 + koff]);   // ds_load_b64
      v2f b1 = *(const v2f*)(&sW[cur][col1][k4 + koff]);   // ds_load_b64
      // 8 args: (neg_a, A, neg_b, B, c_mod, C, reuse_a, reuse_b)
      acc0 = __builtin_amdgcn_wmma_f32_16x16x4_f32(false, a, false, b0,
                                                   (short)0, acc0, false, false);
      acc1 = __builtin_amdgcn_wmma_f32_16x16x4_f32(false, a, false, b1,
                                                   (short)0, acc1, false, false);
    }
  }

  // C/D layout: VGPR r -> lanes 0-15: (M=r, N=lane), lanes 16-31: (M=8+r, N=lane-16)
  const float bias0 = bias[col0];
  const float bias1 = (col1 < TT) ? bias[col1] : 0.0f;
#pragma unroll
  for (int r = 0; r < 8; ++r) {
    const int row = m0 + r + half * 8;
    emission[(size_t)row * TT + col0] = acc0[r] + bias0;
    if (col1 < TT)
      emission[(size_t)row * TT + col1] = acc1[r] + bias1;
  }
}

// ---------------------------------------------------------------------------
// Kernel 2: per-batch CRF forward recursion + gold-path score.
// One wave32 per batch element; lane t owns state t (t < 24 active).
// ---------------------------------------------------------------------------
__global__ __launch_bounds__(32) void crf_forward_kernel(
    const float* __restrict__ em, const int* __restrict__ target,
    const unsigned char* __restrict__ mask, const float* __restrict__ trans,
    float* __restrict__ ws) {
  const int b = blockIdx.x;
  const int t = threadIdx.x;            // 0..31
  const float NEG = -1e30f;

  float tr[TT];
#pragma unroll
  for (int j = 0; j < TT; ++j)
    tr[j] = (t < TT) ? trans[j * TT + t] : 0.0f;

  // --- gold path score (parallel over s, wave reduce) ---
  float sc = 0.0f;
  for (int s = t; s < SS; s += 32) {
    if (mask[b * SS + s]) {
      const int tg = target[b * SS + s];
      float v = em[((size_t)(b * SS + s)) * TT + tg];
      if (s > 0) v += trans[target[b * SS + s - 1] * TT + tg];
      sc += v;
    }
  }
#pragma unroll
  for (int off = 16; off > 0; off >>= 1) sc += __shfl_down(sc, off, 32);
  sc = __shfl(sc, 0, 32);

  // --- forward recursion ---
  float p = (t < TT) ? em[((size_t)(b * SS)) * TT + t] : NEG;
  for (int s = 1; s < SS; ++s) {
    const float emit = (t < TT) ? em[((size_t)(b * SS + s)) * TT + t] : NEG;
    float x[TT];
    float m = NEG;
#pragma unroll
    for (int j = 0; j < TT; ++j) {
      const float pj = __shfl(p, j, 32);
      x[j] = pj + tr[j];
      m = fmaxf(m, x[j]);
    }
    float ssum = 0.0f;
#pragma unroll
    for (int j = 0; j < TT; ++j) ssum += __expf(x[j] - m);
    const float np = m + __logf(ssum) + emit;
    const bool mi = mask[b * SS + s] != 0;
    p = mi ? np : p;
    if (t >= TT) p = NEG;
  }

  // --- log_Z = logsumexp over states ---
  float m = p;
#pragma unroll
  for (int off = 16; off > 0; off >>= 1) m = fmaxf(m, __shfl_down(m, off, 32));
  m = __shfl(m, 0, 32);
  float e = (t < TT) ? __expf(p - m) : 0.0f;
#pragma unroll
  for (int off = 16; off > 0; off >>= 1) e += __shfl_down(e, off, 32);
  if (t == 0) ws[b] = sc - (m + __logf(e));
}

// ---------------------------------------------------------------------------
// Kernel 3: loss = -mean_b(score_b - logZ_b)
// ---------------------------------------------------------------------------
__global__ __launch_bounds__(32) void crf_loss_kernel(const float* __restrict__ ws,
                                                      float* __restrict__ out) {
  const int t = threadIdx.x;            // 0..31
  float v = ws[t] + ws[t + 32];
#pragma unroll
  for (int off = 16; off > 0; off >>= 1) v += __shfl_down(v, off, 32);
  if (t == 0) out[0] = -v / (float)BB;
}

extern "C" void kernel_launch(void* const* d_in, const int* in_sizes, int n_in,
                              void* d_out, int out_size, void* d_ws, size_t ws_size,
                              hipStream_t stream) {
  const float* feats        = (const float*)d_in[0];
  const int* target         = (const int*)d_in[1];
  const unsigned char* mask = (const unsigned char*)d_in[2]; // jnp bool_ = 1 byte
  const float* W            = (const float*)d_in[3];
  const float* bias         = (const float*)d_in[4];
  const float* trans        = (const float*)d_in[5];

  float* out      = (float*)d_out;
  float* emission = out + 1;            // d_out = [loss, emission(b,s,t)]
  float* partial  = (float*)d_ws;       // 64 floats

  // 32768 rows / (8 waves * 16 rows) = 256 blocks
  crf_emission_wmma<<<(BB * SS) / 128, 256, 0, stream>>>(feats, W, bias, emission);
  crf_forward_kernel<<<BB, 32, 0, stream>>>(emission, target, mask, trans, partial);
  crf_loss_kernel<<<1, 32, 0, stream>>>(partial, out);
}